// CenterNetLoss_31147102830885
// MI455X (gfx1250) — compile-verified
//
#include <hip/hip_runtime.h>
#include <hip/hip_bf16.h>

// CenterNet loss, fused single-pass for MI455X (gfx1250, wave32).
//
// Roofline: only dense read is the 16.8MB heatmap; offsets/log_flux are
// gathered at ~2048 center pixels. Bandwidth-bound (~1-3us at 23.3 TB/s).
// Gaussian target rendering is done on-the-fly per 32x32 tile with LDS
// centroid culling; pixel-vs-centroid squared distances are computed with
// V_WMMA_F32_16X16X4_F32 via the rank-4 distance factorization.

#define IMG_H 512
#define IMG_W 512
#define NPTS  128
#define RAD   7          // int(3*sigma+1), sigma=2
#define INV2S2 0.125f    // 1/(2*sigma^2)
#define FEPS  1e-6f

typedef float v2f __attribute__((ext_vector_type(2)));
typedef float v8f __attribute__((ext_vector_type(8)));

// ---------------------------------------------------------------- init ----
__global__ void cnl_init_ws(float* ws) {
    if (threadIdx.x < 8) ws[threadIdx.x] = 0.0f;
}

// ------------------------------------------------- heatmap focal kernel ---
// grid (W/32, H/32, B), block 256 (8 waves). Accumulates:
//   ws[0] += pos focal loss sum, ws[1] += neg focal loss sum,
//   ws[2] += count(target == 1)
__global__ __launch_bounds__(256) void cnl_render_focal(
    const float* __restrict__ heatmap,
    const float* __restrict__ centroids,
    float* __restrict__ ws)
{
    __shared__ float s_cx[NPTS + 1], s_cy[NPTS + 1], s_c2[NPTS + 1];
    __shared__ int   s_cxi[NPTS + 1], s_cyi[NPTS + 1];
    __shared__ int   s_list[NPTS + 16];
    __shared__ int   s_cnt;
    __shared__ float s_acc[3];

    const int b   = blockIdx.z;
    const int tx0 = blockIdx.x * 32;
    const int ty0 = blockIdx.y * 32;
    const int tid = threadIdx.x;

    if (tid == 0) s_cnt = 0;
    if (tid < 3) s_acc[tid] = 0.0f;

    // Load batch centroids, precompute splat parameters.
    for (int n = tid; n < NPTS; n += 256) {
        float ux = centroids[(b * NPTS + n) * 2 + 0];
        float uy = centroids[(b * NPTS + n) * 2 + 1];
        float cx = ux * (float)(IMG_W - 1);
        float cy = uy * (float)(IMG_H - 1);
        int cxi = (int)fminf(fmaxf(rintf(cx), 0.0f), (float)(IMG_W - 1));
        int cyi = (int)fminf(fmaxf(rintf(cy), 0.0f), (float)(IMG_H - 1));
        s_cx[n] = cx; s_cy[n] = cy; s_c2[n] = cx * cx + cy * cy;
        s_cxi[n] = cxi; s_cyi[n] = cyi;
    }
    if (tid == 0) {  // sentinel for shortlist padding: far away + invalid window
        s_cx[NPTS] = -1.0e6f; s_cy[NPTS] = -1.0e6f; s_c2[NPTS] = 2.0e12f;
        s_cxi[NPTS] = -100000; s_cyi[NPTS] = -100000;
    }
    __syncthreads();

    // Cull centroids whose 15x15 window intersects this tile.
    for (int n = tid; n < NPTS; n += 256) {
        int cxi = s_cxi[n], cyi = s_cyi[n];
        if (cxi >= tx0 - RAD && cxi <= tx0 + 31 + RAD &&
            cyi >= ty0 - RAD && cyi <= ty0 + 31 + RAD) {
            int p = atomicAdd(&s_cnt, 1);
            s_list[p] = n;
        }
    }
    __syncthreads();
    const int S    = s_cnt;
    const int Spad = (S + 15) & ~15;
    for (int p = S + tid; p < Spad; p += 256) s_list[p] = NPTS;
    __syncthreads();

    const int lane = tid & 31;
    const int wv   = tid >> 5;
    const int half = lane >> 4;   // D rows 0..7 (half=0) or 8..15 (half=1)
    const int mcol = lane & 15;   // A row / B,D column index

    float pos_part = 0.0f, neg_part = 0.0f, np_part = 0.0f;

    // Each wave: 4 tile rows x 2 groups of 16 pixels.
    for (int ry = 0; ry < 4; ++ry) {
        const int y  = ty0 + wv * 4 + ry;
        const float py = (float)y;
        // Prefetch next heatmap row segment into caches (global_prefetch_b8).
        if (lane == 0)
            __builtin_prefetch(&heatmap[((size_t)b * IMG_H + y) * IMG_W + tx0], 0, 1);

        for (int gx = 0; gx < 2; ++gx) {
            const int x0 = tx0 + gx * 16;

            // A (16x4 f32): row m = [-2px, -2py, px^2+py^2, 1]
            // lanes 0-15 hold K=0,1 ; lanes 16-31 hold K=2,3 (ISA layout).
            float pxm = (float)(x0 + mcol);
            v2f a;
            a.x = half ? (pxm * pxm + py * py) : (-2.0f * pxm);
            a.y = half ? 1.0f                  : (-2.0f * py);

            float vmax[8] = {0, 0, 0, 0, 0, 0, 0, 0};

            for (int c0 = 0; c0 < Spad; c0 += 16) {
                int idx  = s_list[c0 + mcol];
                float cx = s_cx[idx], cy = s_cy[idx];
                int cxi  = s_cxi[idx], cyi = s_cyi[idx];

                // B (4x16 f32): col n = [cx, cy, 1, cx^2+cy^2]^T
                v2f bb;
                bb.x = half ? 1.0f      : cx;
                bb.y = half ? s_c2[idx] : cy;

                v8f cz = {};
                // d2[M][N] = px^2+py^2 - 2(px*cx + py*cy) + cx^2+cy^2
                v8f d = __builtin_amdgcn_wmma_f32_16x16x4_f32(
                    false, a, false, bb, (short)0, cz, false, false);

                bool vy = (y >= cyi - RAD) && (y <= cyi + RAD);
#pragma unroll
                for (int k = 0; k < 8; ++k) {
                    int pxk = x0 + k + half * 8;          // D row M = k + 8*half
                    bool v  = vy && (pxk >= cxi - RAD) && (pxk <= cxi + RAD);
                    float g = v ? __expf(d[k] * (-INV2S2)) : 0.0f;
                    vmax[k] = fmaxf(vmax[k], g);
                }
            }

            // Max over the 16 centroid columns (lanes within each 16-half).
#pragma unroll
            for (int k = 0; k < 8; ++k) {
                float v = vmax[k];
                v = fmaxf(v, __shfl_xor(v, 1, 32));
                v = fmaxf(v, __shfl_xor(v, 2, 32));
                v = fmaxf(v, __shfl_xor(v, 4, 32));
                v = fmaxf(v, __shfl_xor(v, 8, 32));
                vmax[k] = v;
            }

            // 16 active lanes (mcol<8) each evaluate focal loss for 1 pixel.
            if (mcol < 8) {
                int k = lane & 7;
                float t = vmax[0];
#pragma unroll
                for (int j = 1; j < 8; ++j) t = (k == j) ? vmax[j] : t;
                int x = x0 + k + half * 8;
                float pred = heatmap[((size_t)b * IMG_H + y) * IMG_W + x];
                float p = fminf(fmaxf(pred, FEPS), 1.0f - FEPS);
                if (t == 1.0f) {
                    float om = 1.0f - p;
                    pos_part -= om * om * __logf(p);
                    np_part  += 1.0f;
                } else {
                    float omt = 1.0f - t;
                    float w = omt * omt; w = w * w;     // (1-t)^4
                    neg_part -= w * p * p * __logf(1.0f - p);
                }
            }
        }
    }

    atomicAdd(&s_acc[0], pos_part);
    atomicAdd(&s_acc[1], neg_part);
    atomicAdd(&s_acc[2], np_part);
    __syncthreads();
    if (tid == 0) {
        unsafeAtomicAdd(&ws[0], s_acc[0]);
        unsafeAtomicAdd(&ws[1], s_acc[1]);
        unsafeAtomicAdd(&ws[2], s_acc[2]);
    }
}

// ------------------------------------------- center (offset/flux) kernel --
// grid B, block NPTS. Exact "last scatter wins" collision handling.
//   ws[3] += sum |offset - off_t| at centers, ws[4] += sum |lf - flux_t|,
//   ws[5] += number of distinct center pixels.
__global__ __launch_bounds__(NPTS) void cnl_centers(
    const float* __restrict__ offset,
    const float* __restrict__ log_flux,
    const float* __restrict__ centroids,
    const float* __restrict__ gt_flux,
    float* __restrict__ ws)
{
    __shared__ int s_pid[NPTS];
    const int b = blockIdx.x;
    const int n = threadIdx.x;

    float ux = centroids[(b * NPTS + n) * 2 + 0];
    float uy = centroids[(b * NPTS + n) * 2 + 1];
    float cx = ux * (float)(IMG_W - 1);
    float cy = uy * (float)(IMG_H - 1);
    int cxi = (int)fminf(fmaxf(rintf(cx), 0.0f), (float)(IMG_W - 1));
    int cyi = (int)fminf(fmaxf(rintf(cy), 0.0f), (float)(IMG_H - 1));
    float dx = cx - (float)cxi;
    float dy = cy - (float)cyi;
    int pid = cyi * IMG_W + cxi;
    s_pid[n] = pid;
    __syncthreads();

    bool win = true;  // later index overwrites earlier in scatter .set
    for (int m = n + 1; m < NPTS; ++m) win = win && (s_pid[m] != pid);

    if (win) {
        float ox = offset[(((size_t)b * 2 + 0) * IMG_H + cyi) * IMG_W + cxi];
        float oy = offset[(((size_t)b * 2 + 1) * IMG_H + cyi) * IMG_W + cxi];
        float lf = log_flux[((size_t)b * IMG_H + cyi) * IMG_W + cxi];
        unsafeAtomicAdd(&ws[3], fabsf(ox - dx) + fabsf(oy - dy));
        unsafeAtomicAdd(&ws[4], fabsf(lf - gt_flux[b * NPTS + n]));
        unsafeAtomicAdd(&ws[5], 1.0f);
    }
}

// ------------------------------------------------------------- finalize ---
__global__ void cnl_finalize(const float* __restrict__ ws, float* __restrict__ out) {
    if (threadIdx.x == 0 && blockIdx.x == 0) {
        float nphm  = fmaxf(ws[2], 1.0f);
        float l_hm  = (ws[0] + ws[1]) / nphm;          // LHM = 1.0
        float np    = fmaxf(ws[5], 1.0f);
        float l_off = ws[3] / np;                      // LOFF = 1.0
        float l_fl  = 0.1f * ws[4] / np;               // LFLUX = 0.1
        out[0] = l_hm + l_off + l_fl;
        out[1] = l_hm;
        out[2] = l_off;
        out[3] = l_fl;
        out[4] = (float)NPTS;
    }
}

// --------------------------------------------------------------- launch ---
extern "C" void kernel_launch(void* const* d_in, const int* in_sizes, int n_in,
                              void* d_out, int out_size, void* d_ws, size_t ws_size,
                              hipStream_t stream) {
    const float* heatmap   = (const float*)d_in[0];
    const float* offset    = (const float*)d_in[1];
    const float* log_flux  = (const float*)d_in[2];
    const float* centroids = (const float*)d_in[3];
    const float* gt_flux   = (const float*)d_in[4];
    float* out = (float*)d_out;
    float* ws  = (float*)d_ws;

    cnl_init_ws<<<1, 32, 0, stream>>>(ws);
    cnl_render_focal<<<dim3(IMG_W / 32, IMG_H / 32, 16), 256, 0, stream>>>(
        heatmap, centroids, ws);
    cnl_centers<<<16, NPTS, 0, stream>>>(offset, log_flux, centroids, gt_flux, ws);
    cnl_finalize<<<1, 1, 0, stream>>>(ws, out);
}